// DWT2D_Haar_43954695308070
// MI455X (gfx1250) — compile-verified
//
#include <hip/hip_runtime.h>
#include <stdint.h>

// Native clang vector type: required by __builtin_nontemporal_store and gives
// clean b128 LDS loads / global stores.
typedef float v4f __attribute__((ext_vector_type(4)));

// ---- CDNA5 async global->LDS copy path (ASYNCcnt-tracked) ------------------
// IOFFSET is added to BOTH the LDS and global address (CDNA5 ISA),
// so one lds-addr VGPR + one 64-bit global-addr VGPR pair serves all 8 issues.
#define ASYNC_B128(ldsoff, gptr, imm)                                          \
  asm volatile("global_load_async_to_lds_b128 %0, %1, off offset:" #imm        \
               :: "v"(ldsoff), "v"(gptr) : "memory")

#define WAIT_ASYNCCNT(n)                                                       \
  asm volatile("s_wait_asynccnt " #n ::: "memory")

namespace {
constexpr int B = 8, C = 64, H = 512, W = 512;
constexpr int H2 = H / 2, W2 = W / 2;             // 256 x 256 per subband
constexpr int NTILES = B * C * H2;                // 131072 row-pair tiles
constexpr int TILE_FLOATS = 2 * W;                // rows 2h,2h+1: 1024 floats
constexpr int TILE_BYTES  = TILE_FLOATS * 4;      // 4096 B, contiguous in x
constexpr int WAVES_PER_BLOCK = 8;
constexpr int BLOCKS = 2048;
constexpr int TPW = NTILES / (BLOCKS * WAVES_PER_BLOCK);  // 8 tiles per wave
}

__global__ __launch_bounds__(256) void haar_dwt2d_async(const float* __restrict__ x,
                                                        float* __restrict__ out) {
  // 8 waves x 2 stages x 4KiB = 64 KiB (wave-private double buffers; no barriers)
  __shared__ __attribute__((aligned(16))) float smem[WAVES_PER_BLOCK * 2 * TILE_FLOATS];

  const int lane = threadIdx.x & 31;
  const int wave = threadIdx.x >> 5;

  // LDS byte offset of this wave's buffer: flat addr of __shared__ is
  // {SHARED_BASE, offset[31:0]} -> truncation yields the raw LDS offset.
  const unsigned ldsWave =
      (unsigned)(unsigned long long)(const void*)&smem[wave * 2 * TILE_FLOATS];

  const long gw = (long)blockIdx.x * WAVES_PER_BLOCK + wave;
  const int t0 = (int)(gw * TPW);
  if (t0 >= NTILES) return;
  const int tend = (t0 + TPW < NTILES) ? (t0 + TPW) : NTILES;

  auto issue = [&](int tile, int stage) {
    const char* g = (const char*)x + (size_t)tile * TILE_BYTES + (size_t)lane * 16;
    const unsigned lb = ldsWave + (unsigned)stage * TILE_BYTES + (unsigned)lane * 16;
    ASYNC_B128(lb, g, 0);
    ASYNC_B128(lb, g, 512);
    ASYNC_B128(lb, g, 1024);
    ASYNC_B128(lb, g, 1536);
    ASYNC_B128(lb, g, 2048);
    ASYNC_B128(lb, g, 2560);
    ASYNC_B128(lb, g, 3072);
    ASYNC_B128(lb, g, 3584);
  };

  auto compute = [&](int tile, int stage) {
    const int h = tile & (H2 - 1);
    const int c = (tile >> 8) & (C - 1);
    const int b = tile >> 14;

    // Output bases with the reference's interleave+chunk quirk:
    // k = 4c+s; chunk q = k>>6; in-chunk channel j = k&63.
    float* op[4];
#pragma unroll
    for (int s = 0; s < 4; ++s) {
      const int k = 4 * c + s;
      const int q = k >> 6, j = k & 63;
      const size_t ob =
          ((size_t)((q * 8 + b) * 64 + j)) * (size_t)(H2 * W2) + (size_t)h * W2;
      op[s] = out + ob;
    }

    const v4f* sm =
        (const v4f*)&smem[wave * 2 * TILE_FLOATS + stage * TILE_FLOATS];

#pragma unroll
    for (int p = 0; p < 2; ++p) {
      const int f4 = p * 64 + 2 * lane;           // float4 index within row0
      const v4f r0a = sm[f4];                     // row 2h,   cols 2w..2w+3
      const v4f r0b = sm[f4 + 1];                 // row 2h,   cols 2w+4..2w+7
      const v4f r1a = sm[f4 + 128];               // row 2h+1
      const v4f r1b = sm[f4 + 129];

      v4f ll, lh, hl, hh;
#define HAAR_COL(av, bv, cv, dv, comp)                                         \
      { const float s0 = (av) + (bv), d0 = (av) - (bv);                        \
        const float s1 = (cv) + (dv), d1 = (cv) - (dv);                        \
        ll.comp = 0.5f * (s0 + s1); lh.comp = 0.5f * (d0 + d1);                \
        hl.comp = 0.5f * (s0 - s1); hh.comp = 0.5f * (d0 - d1); }
      HAAR_COL(r0a.x, r0a.y, r1a.x, r1a.y, x)
      HAAR_COL(r0a.z, r0a.w, r1a.z, r1a.w, y)
      HAAR_COL(r0b.x, r0b.y, r1b.x, r1b.y, z)
      HAAR_COL(r0b.z, r0b.w, r1b.z, r1b.w, w)
#undef HAAR_COL

      const int col = p * 128 + 4 * lane;         // fully coalesced b128 stores
      __builtin_nontemporal_store(ll, (v4f*)(op[0] + col));
      __builtin_nontemporal_store(lh, (v4f*)(op[1] + col));
      __builtin_nontemporal_store(hl, (v4f*)(op[2] + col));
      __builtin_nontemporal_store(hh, (v4f*)(op[3] + col));
    }
  };

  // Double-buffered async pipeline over this wave's tiles.
  issue(t0, 0);
  for (int t = t0; t < tend; ++t) {
    const int stage = (t - t0) & 1;
    if (t + 1 < tend) {
      issue(t + 1, stage ^ 1);
      WAIT_ASYNCCNT(8);   // oldest 8 (current stage) complete; next 8 in flight
    } else {
      WAIT_ASYNCCNT(0);
    }
    compute(t, stage);
  }
}

extern "C" void kernel_launch(void* const* d_in, const int* in_sizes, int n_in,
                              void* d_out, int out_size, void* d_ws, size_t ws_size,
                              hipStream_t stream) {
  const float* x = (const float*)d_in[0];
  float* out = (float*)d_out;
  (void)in_sizes; (void)n_in; (void)out_size; (void)d_ws; (void)ws_size;
  haar_dwt2d_async<<<dim3(BLOCKS), dim3(256), 0, stream>>>(x, out);
}